// MPNEncoder_377957122116
// MI455X (gfx1250) — compile-verified
//
#include <hip/hip_runtime.h>
#include <math.h>

// ---------------- problem dimensions ----------------
#define H_DIM   300        // hidden
#define HPAD    304        // H padded to /16
#define MMOL    256
#define LATOM   128
#define AROWS   32769      // 1 + M*L
#define NBMAX   6
#define BROWS   131073     // 1 + 4*M*L
#define AFDIM   133
#define BFDIM   147
#define H3      900
#define H3PAD   912
#define MLROWS  32768      // M*L

#define GF_RELU  1
#define GF_ACCUM 2

typedef __attribute__((ext_vector_type(2))) float v2f;
typedef __attribute__((ext_vector_type(8))) float v8f;

// ---------------------------------------------------------------------------
// f32 GEMM on v_wmma_f32_16x16x4_f32 (exact f32 vs. f32 reference).
//   C[m][n] = (ACCUM ? C : 0) + sum_k X[m][k] * Wt[k][n]  (+bias[n]) (+R[m][n])
// Each wave owns a 32x64 macro-tile: 2 M-tiles x 4 N-tiles = 8 v8f
// accumulators. Per K-step(4): 2 A-fragments (contiguous pair -> b64 load,
// reused over 4 N-tiles) + 4 B-fragments (reused over 2 M-tiles) feed 8
// WMMAs -> 1.5 dwords of load per WMMA. B walks with bp += 4*ldw (no per-
// iteration 64-bit index rebuild). Partial N-quads are clamped in-bounds and
// discarded by uniform store guards so EXEC is all-1s at every WMMA.
// A-matrix ISA layout (16x4 f32): lane 0..15 -> M=lane, VGPR{0,1}=K{0,1};
// lanes 16..31 -> K{2,3}. B mirrored. C/D: VGPR r -> M=r+(lane>=16?8:0).
// ---------------------------------------------------------------------------
__global__ __launch_bounds__(128)
void gemm_f32_wmma(const float* __restrict__ X, int ldx, int Mreal, int Kreal,
                   const float* __restrict__ Wt, int ldw,
                   float* __restrict__ C, int ldc,
                   const float* __restrict__ R, int ldr,
                   const float* __restrict__ bias, int bias_len,
                   int Ntiles, int flags)
{
  const int lane = threadIdx.x & 31;
  const int wave = threadIdx.x >> 5;
  const int Nquads = (Ntiles + 3) >> 2;
  const int quad = blockIdx.y * 4 + wave;
  if (quad >= Nquads) return;               // uniform per wave
  const int lmod = lane & 15;
  const int lhi  = lane >> 4;               // 0 or 1 (half-wave)
  const int m0 = blockIdx.x * 32;           // two 16-row tiles
  const int n0 = quad * 64;                 // four 16-col tiles

  // A row pointers, clamped (clamped rows never stored)
  int ar0 = m0 + lmod;      if (ar0 >= Mreal) ar0 = Mreal - 1;
  int ar1 = m0 + 16 + lmod; if (ar1 >= Mreal) ar1 = Mreal - 1;
  const float* __restrict__ xrow0 = X + (size_t)ar0 * ldx;
  const float* __restrict__ xrow1 = X + (size_t)ar1 * ldx;

  bool nvalid[4];
  int ncol[4];
#pragma unroll
  for (int j = 0; j < 4; ++j) {
    nvalid[j] = (quad * 4 + j) < Ntiles;
    ncol[j] = nvalid[j] ? (n0 + j * 16 + lmod) : lmod;  // clamp OOB tiles in-bounds
  }

  v8f acc[2][4];
  if (flags & GF_ACCUM) {
#pragma unroll
    for (int mt = 0; mt < 2; ++mt)
#pragma unroll
      for (int j = 0; j < 4; ++j)
#pragma unroll
        for (int r = 0; r < 8; ++r) {
          int row = m0 + mt * 16 + r + lhi * 8;
          if (row >= Mreal) row = Mreal - 1;
          acc[mt][j][r] = C[(size_t)row * ldc + ncol[j]];
        }
  } else {
#pragma unroll
    for (int mt = 0; mt < 2; ++mt)
#pragma unroll
      for (int j = 0; j < 4; ++j)
#pragma unroll
        for (int r = 0; r < 8; ++r) acc[mt][j][r] = 0.0f;
  }

  const float* __restrict__ bp = Wt + (size_t)(lhi * 2) * ldw;
  const int kmain = Kreal & ~3;
  for (int kb = 0; kb < kmain; kb += 4) {
    const int k0 = kb + lhi * 2;
    v2f a0, a1;
    a0.x = xrow0[k0]; a0.y = xrow0[k0 + 1];
    a1.x = xrow1[k0]; a1.y = xrow1[k0 + 1];
    v2f b[4];
#pragma unroll
    for (int j = 0; j < 4; ++j) {
      b[j].x = bp[ncol[j]];
      b[j].y = bp[(size_t)ldw + ncol[j]];
    }
#pragma unroll
    for (int j = 0; j < 4; ++j) {
      acc[0][j] = __builtin_amdgcn_wmma_f32_16x16x4_f32(false, a0, false, b[j],
                                                        (short)0, acc[0][j], false, false);
      acc[1][j] = __builtin_amdgcn_wmma_f32_16x16x4_f32(false, a1, false, b[j],
                                                        (short)0, acc[1][j], false, false);
    }
    bp += (size_t)4 * ldw;
  }
  if (kmain < Kreal) {                      // K tail (K=133,147); Wt rows zero-padded
    const int k0 = kmain + lhi * 2;
    v2f a0, a1, b[4];
    a0.x = (k0     < Kreal) ? xrow0[k0]     : 0.0f;
    a0.y = (k0 + 1 < Kreal) ? xrow0[k0 + 1] : 0.0f;
    a1.x = (k0     < Kreal) ? xrow1[k0]     : 0.0f;
    a1.y = (k0 + 1 < Kreal) ? xrow1[k0 + 1] : 0.0f;
#pragma unroll
    for (int j = 0; j < 4; ++j) {
      b[j].x = bp[ncol[j]];
      b[j].y = bp[(size_t)ldw + ncol[j]];
    }
#pragma unroll
    for (int j = 0; j < 4; ++j) {
      acc[0][j] = __builtin_amdgcn_wmma_f32_16x16x4_f32(false, a0, false, b[j],
                                                        (short)0, acc[0][j], false, false);
      acc[1][j] = __builtin_amdgcn_wmma_f32_16x16x4_f32(false, a1, false, b[j],
                                                        (short)0, acc[1][j], false, false);
    }
  }

  float bv[4];
#pragma unroll
  for (int j = 0; j < 4; ++j)
    bv[j] = (bias && ncol[j] < bias_len) ? bias[ncol[j]] : 0.0f;

#pragma unroll
  for (int mt = 0; mt < 2; ++mt)
#pragma unroll
    for (int j = 0; j < 4; ++j) {
      if (!nvalid[j]) continue;             // uniform per wave
#pragma unroll
      for (int r = 0; r < 8; ++r) {
        int row = m0 + mt * 16 + r + lhi * 8;
        float v = acc[mt][j][r] + bv[j];
        if (R) {
          int rr = (row < Mreal) ? row : 0;
          v += R[(size_t)rr * ldr + ncol[j]];
        }
        if (flags & GF_RELU) v = fmaxf(v, 0.0f);
        if (row < Mreal) C[(size_t)row * ldc + ncol[j]] = v;
      }
    }
}

// Wt[k][n] = (k<Kin && n<Nout) ? W[n][k] : 0, shape [Kp x Np]
__global__ void transpose_pad(const float* __restrict__ W, int Nout, int Kin,
                              float* __restrict__ Wt, int Np, int Kp)
{
  int idx = blockIdx.x * blockDim.x + threadIdx.x;
  if (idx >= Kp * Np) return;
  int k = idx / Np, n = idx % Np;
  Wt[idx] = (k < Kin && n < Nout) ? W[(size_t)n * Kin + k] : 0.0f;
}

// agg[a][h] = (sum_j mb[a2b[a][j]][h]) * (max_j mb[a2b[a][j]][h])
__global__ void aggregate_kernel(const float* __restrict__ mb,
                                 const int* __restrict__ a2b,
                                 float* __restrict__ out, int addInPlace)
{
  int a = blockIdx.x, h = threadIdx.x;
  if (h >= H_DIM) return;
  float s = 0.0f, mx = -3.402823466e+38f;
  for (int j = 0; j < NBMAX; ++j) {
    int idx = a2b[a * NBMAX + j];
    float v = mb[(size_t)idx * HPAD + h];
    s += v;
    mx = fmaxf(mx, v);
  }
  float agg = s * mx;
  if (addInPlace) out[(size_t)a * HPAD + h] += agg;
  else            out[(size_t)a * HPAD + h]  = agg;
}

// tmp[b][h] = message_atom[b2a[b]][h] - mb[b2revb[b]][h]
__global__ void bond_update_kernel(const float* __restrict__ ma,
                                   const float* __restrict__ mb,
                                   const int* __restrict__ b2a,
                                   const int* __restrict__ b2revb,
                                   float* __restrict__ tmp)
{
  int b = blockIdx.x, h = threadIdx.x;
  if (h >= H_DIM) return;
  int a = b2a[b], rb = b2revb[b];
  tmp[(size_t)b * HPAD + h] = ma[(size_t)a * HPAD + h] - mb[(size_t)rb * HPAD + h];
}

// msg = relu(node + gru_bias)
__global__ void message_relu_kernel(const float* __restrict__ node,
                                    const float* __restrict__ gbias,
                                    float* __restrict__ msg)
{
  int r = blockIdx.x, h = threadIdx.x;
  if (h >= H_DIM) return;
  msg[(size_t)r * HPAD + h] = fmaxf(node[(size_t)r * HPAD + h] + gbias[h], 0.0f);
}

// h0[m][k] = max_t node[1 + m*L + t][k], broadcast to forward/backward states
__global__ void h0_kernel(const float* __restrict__ node,
                          float* __restrict__ hf, float* __restrict__ hb)
{
  int m = blockIdx.x, k = threadIdx.x;
  if (k >= H_DIM) return;
  float mx = -3.402823466e+38f;
  for (int t = 0; t < LATOM; ++t)
    mx = fmaxf(mx, node[(size_t)(1 + m * LATOM + t) * HPAD + k]);
  hf[(size_t)m * HPAD + k] = mx;
  hb[(size_t)m * HPAD + k] = mx;
}

// torch GRU gate math; gi/gh already include bih/bhh
__global__ void gru_gate_kernel(const float* __restrict__ gi,  // [ML, H3P]
                                const float* __restrict__ gh,  // [M,  H3P]
                                float* __restrict__ h,         // [M,  HPAD]
                                float* __restrict__ out,       // [ML, HPAD]
                                int t)
{
  int m = blockIdx.x, i = threadIdx.x;
  if (i >= H_DIM) return;
  size_t grow = (size_t)m * LATOM + t;
  float ir = gi[grow * H3PAD + i];
  float iz = gi[grow * H3PAD + H_DIM + i];
  float in = gi[grow * H3PAD + 2 * H_DIM + i];
  float hr = gh[(size_t)m * H3PAD + i];
  float hz = gh[(size_t)m * H3PAD + H_DIM + i];
  float hn = gh[(size_t)m * H3PAD + 2 * H_DIM + i];
  float r = 1.0f / (1.0f + expf(-(ir + hr)));
  float z = 1.0f / (1.0f + expf(-(iz + hz)));
  float n = tanhf(in + r * hn);
  float hv = h[(size_t)m * HPAD + i];
  float hnew = (1.0f - z) * n + z * hv;
  h[(size_t)m * HPAD + i] = hnew;
  out[grow * HPAD + i] = hnew;
}

// mol_vecs[m][h] = mean_t ah[m*L+t][h]
__global__ void mean_kernel(const float* __restrict__ ah, float* __restrict__ out)
{
  int m = blockIdx.x, h = threadIdx.x;
  if (h >= H_DIM) return;
  float s = 0.0f;
  for (int t = 0; t < LATOM; ++t)
    s += ah[(size_t)(m * LATOM + t) * HPAD + h];
  out[(size_t)m * H_DIM + h] = s * (1.0f / LATOM);
}

// ---------------------------------------------------------------------------
static void gemm(hipStream_t s, const float* X, int ldx, int Mreal, int Kreal,
                 const float* Wt, int Np, float* C, int ldc,
                 const float* R, int ldr, const float* bias, int blen, int flags)
{
  int Ntiles = Np / 16;
  int Nquads = (Ntiles + 3) / 4;
  dim3 grid((Mreal + 31) / 32, (Nquads + 3) / 4);
  gemm_f32_wmma<<<grid, 128, 0, s>>>(X, ldx, Mreal, Kreal, Wt, Np, C, ldc,
                                     R, ldr, bias, blen, Ntiles, flags);
}

extern "C" void kernel_launch(void* const* d_in, const int* in_sizes, int n_in,
                              void* d_out, int out_size, void* d_ws, size_t ws_size,
                              hipStream_t stream)
{
  const float* f_atoms  = (const float*)d_in[0];
  const float* f_bonds  = (const float*)d_in[1];
  const int*   a2b      = (const int*)d_in[2];
  const int*   b2a      = (const int*)d_in[3];
  const int*   b2revb   = (const int*)d_in[4];
  const float* W_i_atom = (const float*)d_in[5];
  const float* W_i_bond = (const float*)d_in[6];
  const float* W_h0     = (const float*)d_in[7];
  const float* W_h1     = (const float*)d_in[8];
  const float* W_lr     = (const float*)d_in[9];
  const float* gru_bias = (const float*)d_in[10];
  const float* Wih_f    = (const float*)d_in[11];
  const float* Whh_f    = (const float*)d_in[12];
  const float* bih_f    = (const float*)d_in[13];
  const float* bhh_f    = (const float*)d_in[14];
  const float* Wih_b    = (const float*)d_in[15];
  const float* Whh_b    = (const float*)d_in[16];
  const float* bih_b    = (const float*)d_in[17];
  const float* bhh_b    = (const float*)d_in[18];
  const float* W_o      = (const float*)d_in[19];
  const float* b_o      = (const float*)d_in[20];
  float* out = (float*)d_out;

  // ---------------- workspace layout (floats), ~650 MB, heavily aliased ----
  float* base = (float*)d_ws;
  size_t o = 0;
  auto alloc = [&](size_t n) { float* p = base + o; o += (n + 63) & ~(size_t)63; return p; };
  const size_t BP = 131088, AP = 32784;            // row-padded counts
  float* IB   = alloc(BP * HPAD);                  // input_bond; later gi_b
  float* MBA  = alloc(BP * HPAD);                  // message_bond; later out_f/out_b/ah
  float* TMP  = alloc(BP * HPAD);                  // bond scratch; later gi_f
  float* IA   = alloc(AP * HPAD);                  // input_atom
  float* MA   = alloc(AP * HPAD);                  // message_atom
  float* AGG  = alloc(AP * HPAD);                  // agg; later message (relu)
  float* NODE = alloc(AP * HPAD);
  float* GH   = alloc((size_t)MMOL * H3PAD);
  float* HF   = alloc((size_t)MMOL * HPAD);
  float* HB   = alloc((size_t)MMOL * HPAD);
  float* WtA  = alloc((size_t)136 * HPAD);
  float* WtB  = alloc((size_t)148 * HPAD);
  float* WtH0 = alloc((size_t)300 * HPAD);
  float* WtH1 = alloc((size_t)300 * HPAD);
  float* WtLR = alloc((size_t)900 * HPAD);
  float* WtIHf= alloc((size_t)300 * H3PAD);
  float* WtHHf= alloc((size_t)300 * H3PAD);
  float* WtIHb= alloc((size_t)300 * H3PAD);
  float* WtHHb= alloc((size_t)300 * H3PAD);
  float* WtO  = alloc((size_t)600 * HPAD);
  // aliases (lifetimes disjoint by stream order):
  float* GIF  = TMP;                               // [ML, H3PAD]
  float* GIB  = IB;                                // [ML, H3PAD]
  float* OUTF = MBA;                               // [ML, HPAD]
  float* OUTB = MBA + (size_t)MLROWS * HPAD;       // [ML, HPAD]
  float* AH   = MBA + (size_t)2 * MLROWS * HPAD;   // [ML, HPAD]

  // ---------------- weight transpose + zero-pad ----------------
  auto tp = [&](const float* W, int Nout, int Kin, float* Wt, int Np, int Kp) {
    int total = Kp * Np;
    transpose_pad<<<(total + 255) / 256, 256, 0, stream>>>(W, Nout, Kin, Wt, Np, Kp);
  };
  tp(W_i_atom, 300, 133, WtA, HPAD, 136);
  tp(W_i_bond, 300, 147, WtB, HPAD, 148);
  tp(W_h0,     300, 300, WtH0, HPAD, 300);
  tp(W_h1,     300, 300, WtH1, HPAD, 300);
  tp(W_lr,     300, 900, WtLR, HPAD, 900);
  tp(Wih_f,    900, 300, WtIHf, H3PAD, 300);
  tp(Whh_f,    900, 300, WtHHf, H3PAD, 300);
  tp(Wih_b,    900, 300, WtIHb, H3PAD, 300);
  tp(Whh_b,    900, 300, WtHHb, H3PAD, 300);
  tp(W_o,      300, 600, WtO, HPAD, 600);

  // ---------------- input projections ----------------
  gemm(stream, f_atoms, AFDIM, AROWS, AFDIM, WtA, HPAD, IA, HPAD,
       nullptr, 0, nullptr, 0, GF_RELU);
  gemm(stream, f_bonds, BFDIM, BROWS, BFDIM, WtB, HPAD, IB, HPAD,
       nullptr, 0, nullptr, 0, GF_RELU);
  hipMemcpyAsync(MA, IA, (size_t)AROWS * HPAD * sizeof(float),
                 hipMemcpyDeviceToDevice, stream);   // message_atom = input_atom

  // ---------------- depth loop (2 iterations) ----------------
  // d = 0: message_bond == input_bond (IB)
  aggregate_kernel<<<AROWS, 320, 0, stream>>>(IB, a2b, MA, 1);
  bond_update_kernel<<<BROWS, 320, 0, stream>>>(MA, IB, b2a, b2revb, TMP);
  gemm(stream, TMP, HPAD, BROWS, H_DIM, WtH0, HPAD, MBA, HPAD,
       IB, HPAD, nullptr, 0, GF_RELU);
  // d = 1
  aggregate_kernel<<<AROWS, 320, 0, stream>>>(MBA, a2b, MA, 1);
  bond_update_kernel<<<BROWS, 320, 0, stream>>>(MA, MBA, b2a, b2revb, TMP);
  gemm(stream, TMP, HPAD, BROWS, H_DIM, WtH1, HPAD, MBA, HPAD,
       IB, HPAD, nullptr, 0, GF_RELU);

  // ---------------- final aggregation + node = cat3 @ W_lr^T --------------
  aggregate_kernel<<<AROWS, 320, 0, stream>>>(MBA, a2b, AGG, 0);
  gemm(stream, AGG, HPAD, AROWS, H_DIM, WtLR,              HPAD, NODE, HPAD,
       nullptr, 0, nullptr, 0, 0);
  gemm(stream, MA,  HPAD, AROWS, H_DIM, WtLR + 300 * HPAD, HPAD, NODE, HPAD,
       nullptr, 0, nullptr, 0, GF_ACCUM);
  gemm(stream, IA,  HPAD, AROWS, H_DIM, WtLR + 600 * HPAD, HPAD, NODE, HPAD,
       nullptr, 0, nullptr, 0, GF_ACCUM);

  // message = relu(node + gru_bias) (into AGG); h0 = max_t node
  message_relu_kernel<<<AROWS, 320, 0, stream>>>(NODE, gru_bias, AGG);
  h0_kernel<<<MMOL, 320, 0, stream>>>(NODE, HF, HB);

  // gi_all = msg @ Wih^T + bih  (rows 1.. of message -> ML rows)
  gemm(stream, AGG + HPAD, HPAD, MLROWS, H_DIM, WtIHf, H3PAD, GIF, H3PAD,
       nullptr, 0, bih_f, H3, 0);
  gemm(stream, AGG + HPAD, HPAD, MLROWS, H_DIM, WtIHb, H3PAD, GIB, H3PAD,
       nullptr, 0, bih_b, H3, 0);

  // ---------------- sequential GRU, forward then backward ----------------
  for (int t = 0; t < LATOM; ++t) {
    gemm(stream, HF, HPAD, MMOL, H_DIM, WtHHf, H3PAD, GH, H3PAD,
         nullptr, 0, bhh_f, H3, 0);
    gru_gate_kernel<<<MMOL, 320, 0, stream>>>(GIF, GH, HF, OUTF, t);
  }
  for (int s = 0; s < LATOM; ++s) {
    int t = LATOM - 1 - s;
    gemm(stream, HB, HPAD, MMOL, H_DIM, WtHHb, H3PAD, GH, H3PAD,
         nullptr, 0, bhh_b, H3, 0);
    gru_gate_kernel<<<MMOL, 320, 0, stream>>>(GIB, GH, HB, OUTB, t);
  }

  // ---------------- atom_hiddens = relu([out_f|out_b] @ W_o^T + b_o) -------
  // (row 0 of agg2 never contributes to mol_vecs -> skipped)
  gemm(stream, OUTF, HPAD, MLROWS, H_DIM, WtO,              HPAD, AH, HPAD,
       nullptr, 0, nullptr, 0, 0);
  gemm(stream, OUTB, HPAD, MLROWS, H_DIM, WtO + 300 * HPAD, HPAD, AH, HPAD,
       nullptr, 0, b_o, H_DIM, GF_ACCUM | GF_RELU);

  // ---------------- per-molecule mean -> d_out [256, 300] ----------------
  mean_kernel<<<MMOL, 320, 0, stream>>>(AH, out);
  (void)in_sizes; (void)n_in; (void)out_size; (void)ws_size;
}